// GraphOUProcess_71854802862393
// MI455X (gfx1250) — compile-verified
//
#include <hip/hip_runtime.h>
#include <hip/hip_bf16.h>

typedef __attribute__((ext_vector_type(2))) float v2f;
typedef __attribute__((ext_vector_type(8))) float v8f;

#define NDIM 512
#define NMAT 32          // g * bs = 2 * 16
#define BS   16
#define GDIM 2
#define TSTEPS 99

// ---------------------------------------------------------------------------
// Kernel 1/2: C[mat] = (A[mat] @ B[mat]) * scale   for 32 independent 512x512
// f32 matrices, using V_WMMA_F32_16X16X4_F32.
// Each wave (wave32) computes a 16(M) x 64(N) strip of C: 4 accumulators,
// A-fragment reused across 4 WMMAs per K-chunk. 8 waves/block, 32 blocks/matrix.
// ---------------------------------------------------------------------------
__global__ void __launch_bounds__(256)
gemm512_wmma(const float* __restrict__ A, const float* __restrict__ B,
             float* __restrict__ C, float scale) {
    const int mat  = blockIdx.y;
    const float* Am = A + (size_t)mat * NDIM * NDIM;
    const float* Bm = B + (size_t)mat * NDIM * NDIM;
    float*       Cm = C + (size_t)mat * NDIM * NDIM;

    const int wave = threadIdx.x >> 5;       // wave32
    const int lane = threadIdx.x & 31;
    const int tid  = blockIdx.x * 8 + wave;  // 0..255 wave-tiles per matrix
    const int M0   = (tid >> 3) * 16;        // 32 M-tiles
    const int N0   = (tid & 7) * 64;         // 8 N-strips

    const int row  = M0 + (lane & 15);
    const int koff = (lane >> 4) * 2;        // lanes 16-31 hold K+2, K+3
    const int col  = lane & 15;

    v8f acc0 = {}, acc1 = {}, acc2 = {}, acc3 = {};

    for (int k = 0; k < NDIM; k += 4) {
        // A fragment: 16x4, 32-bit elements: VGPR0=K(+koff), VGPR1=K(+koff)+1
        const float* ap = Am + (size_t)row * NDIM + k + koff;
        v2f a; a.x = ap[0]; a.y = ap[1];

        // B fragments: 4x16 per sub-tile; VGPR0 = row k+koff, VGPR1 = row k+koff+1
        const float* bp = Bm + (size_t)(k + koff) * NDIM;
        v2f b0, b1, b2, b3;
        b0.x = bp[N0      + col]; b0.y = bp[NDIM + N0      + col];
        b1.x = bp[N0 + 16 + col]; b1.y = bp[NDIM + N0 + 16 + col];
        b2.x = bp[N0 + 32 + col]; b2.y = bp[NDIM + N0 + 32 + col];
        b3.x = bp[N0 + 48 + col]; b3.y = bp[NDIM + N0 + 48 + col];

        acc0 = __builtin_amdgcn_wmma_f32_16x16x4_f32(false, a, false, b0,
                                                     (short)0, acc0, false, false);
        acc1 = __builtin_amdgcn_wmma_f32_16x16x4_f32(false, a, false, b1,
                                                     (short)0, acc1, false, false);
        acc2 = __builtin_amdgcn_wmma_f32_16x16x4_f32(false, a, false, b2,
                                                     (short)0, acc2, false, false);
        acc3 = __builtin_amdgcn_wmma_f32_16x16x4_f32(false, a, false, b3,
                                                     (short)0, acc3, false, false);
    }

    // C/D layout: VGPR r -> rows M0+r (lanes 0-15) / M0+8+r (lanes 16-31)
    const int rbase = M0 + ((lane >> 4) << 3);
#pragma unroll
    for (int r = 0; r < 8; ++r) {
        float* cp = Cm + (size_t)(rbase + r) * NDIM + N0 + col;
        cp[0]  = acc0[r] * scale;
        cp[16] = acc1[r] * scale;
        cp[32] = acc2[r] * scale;
        cp[48] = acc3[r] * scale;
    }
}

// ---------------------------------------------------------------------------
// Kernel 3: deg[g,p,m] = sum_n adj_list[batch0, g, p, n, m]
// Reference computes deg from BATCH 0 ONLY (adj_list[0]); p=0 matrix is
// adj/512 (scaled inline), p=1 -> M2, p=2 -> M3 (already scaled).
// grid = 6 blocks (g*3), block = 512 threads.
// ---------------------------------------------------------------------------
__global__ void __launch_bounds__(512)
deg_kernel(const float* __restrict__ adj, const float* __restrict__ M2,
           const float* __restrict__ M3, float* __restrict__ deg) {
    const int g_i = blockIdx.x / 3;
    const int p   = blockIdx.x % 3;
    const int m   = threadIdx.x;

    const float* M;
    float sc = 1.0f;
    if (p == 0)      { M = adj; sc = 1.0f / (float)NDIM; }
    else if (p == 1) { M = M2; }
    else             { M = M3; }

    const float* base = M + (size_t)(g_i * BS + 0) * NDIM * NDIM;  // batch 0
    float s = 0.0f;
    for (int n = 0; n < NDIM; ++n) s += base[(size_t)n * NDIM + m];
    deg[(g_i * 3 + p) * NDIM + m] = s * sc;
}

// ---------------------------------------------------------------------------
// Kernel 4: the sequential OU scan. 96 independent chains (b, g, p); one
// workgroup per chain, x_last held in LDS, 99 steps. Each thread owns two
// consecutive columns m = 2*tid, 2*tid+1 (float2 coalesced matrix reads).
// Matrices stay L2-resident (96 MB total < 192 MB L2).
// ---------------------------------------------------------------------------
__global__ void __launch_bounds__(256)
ou_scan(const int* __restrict__ skill_seq, const int* __restrict__ time_seq,
        const float* __restrict__ adj, const float* __restrict__ M2,
        const float* __restrict__ M3, const float* __restrict__ speed_in,
        const float* __restrict__ level_in, const float* __restrict__ vola_in,
        const float* __restrict__ noise, const float* __restrict__ deg,
        float* __restrict__ out) {
    const int chain = blockIdx.x;            // 0..95
    const int b     = chain / 6;
    const int g_i   = (chain % 6) / 3;
    const int p     = chain % 3;
    const int tid   = threadIdx.x;
    const int m0    = tid * 2;

    __shared__ float xs[NDIM];
    xs[m0] = 0.0f; xs[m0 + 1] = 0.0f;

    const float* M;
    float msc = 1.0f;
    if (p == 0)      { M = adj; msc = 1.0f / (float)NDIM; }
    else if (p == 1) { M = M2; }
    else             { M = M3; }
    const float* Mb = M + (size_t)(g_i * BS + b) * NDIM * NDIM;

    const float speed = speed_in[b];
    const float level = level_in[b];
    const float vol   = vola_in[b];

    const float* degp = deg + (g_i * 3 + p) * NDIM;
    const float* deg0 = deg + (g_i * 3 + 0) * NDIM;
    const float  da   = degp[m0],     db = degp[m0 + 1];
    const float  za   = deg0[m0],     zb = deg0[m0 + 1];
    const float  inva = 1.0f / (da + 1e-7f);
    const float  invb = 1.0f / (db + 1e-7f);

    float xla = 0.0f, xlb = 0.0f;   // x_last for this thread's two columns
    __syncthreads();

    for (int t = 0; t < TSTEPS; ++t) {
        const float dt = (float)(time_seq[b * 100 + t + 1] -
                                 time_seq[b * 100 + t]);
        const float e   = __expf(-speed * dt);
        const float e2  = __expf(-2.0f * speed * dt);
        const float std = vol * sqrtf((1.0f - e2) / (2.0f * speed));
        const float lvl = (1.0f - e) * level;

        // s[m] = sum_n M[n][m] * x_last[n]  (column dot; row reads coalesced)
        float sa = 0.0f, sb = 0.0f;
        for (int n = 0; n < NDIM; n += 4) {
            const float4 xn = *(const float4*)&xs[n];
            const float* mp = Mb + (size_t)n * NDIM + m0;
            const float2 r0 = *(const float2*)(mp);
            const float2 r1 = *(const float2*)(mp + NDIM);
            const float2 r2 = *(const float2*)(mp + 2 * NDIM);
            const float2 r3 = *(const float2*)(mp + 3 * NDIM);
            sa += r0.x * xn.x + r1.x * xn.y + r2.x * xn.z + r3.x * xn.w;
            sb += r0.y * xn.x + r1.y * xn.y + r2.y * xn.z + r3.y * xn.w;
        }
        sa *= msc; sb *= msc;
        sa = (za == 0.0f) ? 1.0f : sa * inva;
        sb = (zb == 0.0f) ? 1.0f : sb * invb;

        const float* nz =
            noise + ((((size_t)t * BS + b) * GDIM + g_i) * 3 + p) * NDIM;
        const float2 nv = *(const float2*)(nz + m0);

        const float arga = nv.x * std + xla * e + lvl * sa;
        const float argb = nv.y * std + xlb * e + lvl * sb;
        const float xta  = 1.0f / (1.0f + __expf(-arga));
        const float xtb  = 1.0f / (1.0f + __expf(-argb));

        const int skl = skill_seq[b * 100 + t + 1];
        if (skl == m0)
            out[((size_t)t * BS + b) * (GDIM * 3) + g_i * 3 + p] = xta;
        if (skl == m0 + 1)
            out[((size_t)t * BS + b) * (GDIM * 3) + g_i * 3 + p] = xtb;

        __syncthreads();
        xs[m0] = xta; xs[m0 + 1] = xtb;
        xla = xta; xlb = xtb;
        __syncthreads();
    }
}

// ---------------------------------------------------------------------------
// Launch. Inputs (setup_inputs order):
//  0 skill_seq (int 16x100), 1 time_seq (int 16x100), 2 label_seq (unused),
//  3 adj (f32 2x16x512x512), 4 mean_rev_speed (f32 16), 5 mean_rev_level,
//  6 vola, 7 noise (f32 99x16x2x3x512).
// Output: preds f32 (99,16,2,3).
// Workspace: M2 (32 MB) | M3 (32 MB) | deg (12 KB)
// ---------------------------------------------------------------------------
extern "C" void kernel_launch(void* const* d_in, const int* in_sizes, int n_in,
                              void* d_out, int out_size, void* d_ws,
                              size_t ws_size, hipStream_t stream) {
    const int*   skill = (const int*)d_in[0];
    const int*   timev = (const int*)d_in[1];
    const float* adj   = (const float*)d_in[3];
    const float* speed = (const float*)d_in[4];
    const float* level = (const float*)d_in[5];
    const float* vola  = (const float*)d_in[6];
    const float* noise = (const float*)d_in[7];
    float*       out   = (float*)d_out;

    const size_t matElems = (size_t)NMAT * NDIM * NDIM;  // 8,388,608
    float* M2  = (float*)d_ws;
    float* M3  = M2 + matElems;
    float* deg = M3 + matElems;

    const float inv_n  = 1.0f / (float)NDIM;
    const float inv_n2 = inv_n * inv_n;

    dim3 gemmGrid(32, NMAT);
    // M2 = (adj @ adj) / n^2
    gemm512_wmma<<<gemmGrid, 256, 0, stream>>>(adj, adj, M2, inv_n2);
    // M3 = (M2 @ adj) / n   == adj^3 / n^3
    gemm512_wmma<<<gemmGrid, 256, 0, stream>>>(M2, adj, M3, inv_n);
    // deg (batch-0 column sums, matching reference's adj_list[0])
    deg_kernel<<<6, 512, 0, stream>>>(adj, M2, M3, deg);
    // sequential scan: 96 independent chains
    ou_scan<<<96, 256, 0, stream>>>(skill, timev, adj, M2, M3, speed, level,
                                    vola, noise, deg, out);
}